// OuterProductMean_5411658793429
// MI455X (gfx1250) — compile-verified
//
#include <hip/hip_runtime.h>

typedef __attribute__((ext_vector_type(2))) float v2f;
typedef __attribute__((ext_vector_type(8))) float v8f;

#define L_DIM   256
#define IN_DIM  256
#define M_DIM   32
#define P_DIM   128
#define N_COL   4096   // P_DIM * M_DIM

// ---------------------------------------------------------------------------
// Kernel A: X[l][m] = seq[l,:] . W1[m,:] + b1[m]     (256x32 = 256x256 @ 256x32)
// Wave-per-tile, V_WMMA_F32_16X16X4_F32, K=256 in 64 chunks.
// ---------------------------------------------------------------------------
__global__ __launch_bounds__(256) void opm_proj1(const float* __restrict__ seq,
                                                 const float* __restrict__ W1,
                                                 const float* __restrict__ b1,
                                                 float* __restrict__ X) {
    const int lane   = threadIdx.x & 31;
    const int lane15 = lane & 15;
    const int hi     = lane >> 4;
    const int w      = blockIdx.x * 8 + (threadIdx.x >> 5);  // 0..31
    const int m0     = (w >> 1) * 16;                        // l-tile
    const int n0     = (w & 1) * 16;                         // m-tile

    const float* pA = seq + (m0 + lane15) * IN_DIM + 2 * hi;
    const float* pB = W1  + (n0 + lane15) * IN_DIM + 2 * hi;

    v8f acc = {};
#pragma unroll
    for (int kb = 0; kb < IN_DIM; kb += 4) {
        v2f a = *(const v2f*)(pA + kb);
        v2f b = *(const v2f*)(pB + kb);
        acc = __builtin_amdgcn_wmma_f32_16x16x4_f32(false, a, false, b,
                                                    (short)0, acc, false, false);
    }
    const float bb = b1[n0 + lane15];
#pragma unroll
    for (int v = 0; v < 8; ++v)
        X[(m0 + v + 8 * hi) * M_DIM + n0 + lane15] = acc[v] + bb;
}

// ---------------------------------------------------------------------------
// Kernel B: Y[i][n] = sum_d X[i][d] * W2_flat[n*32 + d]   (256x4096 = 256x32 @ 32x4096)
// W2 (P x M*M) row-major is exactly a (4096 x 32) K-major matrix.
// ---------------------------------------------------------------------------
__global__ __launch_bounds__(256) void opm_proj2(const float* __restrict__ X,
                                                 const float* __restrict__ W2,
                                                 float* __restrict__ Y) {
    const int lane   = threadIdx.x & 31;
    const int lane15 = lane & 15;
    const int hi     = lane >> 4;
    const int w      = blockIdx.x * 8 + (threadIdx.x >> 5);  // 0..4095
    const int m0     = (w >> 8) * 16;                        // i-tile (16)
    const int n0     = (w & 255) * 16;                       // n-tile (256)

    const float* pA = X  + (m0 + lane15) * M_DIM + 2 * hi;
    const float* pB = W2 + (n0 + lane15) * M_DIM + 2 * hi;

    v8f acc = {};
#pragma unroll
    for (int kb = 0; kb < M_DIM; kb += 4) {
        v2f a = *(const v2f*)(pA + kb);
        v2f b = *(const v2f*)(pB + kb);
        acc = __builtin_amdgcn_wmma_f32_16x16x4_f32(false, a, false, b,
                                                    (short)0, acc, false, false);
    }
#pragma unroll
    for (int v = 0; v < 8; ++v)
        Y[(m0 + v + 8 * hi) * N_COL + n0 + lane15] = acc[v];
}

// ---------------------------------------------------------------------------
// Kernel C (streaming, bandwidth-bound):
//   out[i,j,p] = sum_c X[j][c] * Y[i][p*32+c] + b2[p] + bias[i,j,p]
// Block = (i, half of j), wave = p-tile. B operand (Y slice) hoisted out of the
// j loop; epilogue fuses b2 + bias with non-temporal load/store (single touch).
// ---------------------------------------------------------------------------
__global__ __launch_bounds__(256) void opm_outer(const float* __restrict__ X,
                                                 const float* __restrict__ Y,
                                                 const float* __restrict__ b2,
                                                 const float* __restrict__ bias,
                                                 float* __restrict__ out) {
    const int lane   = threadIdx.x & 31;
    const int lane15 = lane & 15;
    const int hi     = lane >> 4;
    const int wid    = threadIdx.x >> 5;   // 0..7 -> p-tile
    const int i      = blockIdx.x;         // 0..255
    const int jhalf  = blockIdx.y;         // 0..1
    const int p0     = wid * 16;

    // Preload B = Y[i, (p0+lane15)*32 + c] once per wave (K-major rows).
    const float* pB = Y + i * N_COL + (p0 + lane15) * M_DIM + 2 * hi;
    v2f Bv[8];
#pragma unroll
    for (int c = 0; c < 8; ++c) Bv[c] = *(const v2f*)(pB + 4 * c);

    const float bb = b2[p0 + lane15];

    for (int jt = 0; jt < 8; ++jt) {
        const int j0 = (jhalf * 8 + jt) * 16;
        const float* pA = X + (j0 + lane15) * M_DIM + 2 * hi;

        v8f acc = {};
#pragma unroll
        for (int c = 0; c < 8; ++c) {
            v2f a = *(const v2f*)(pA + 4 * c);
            acc = __builtin_amdgcn_wmma_f32_16x16x4_f32(false, a, false, Bv[c],
                                                        (short)0, acc, false, false);
        }
#pragma unroll
        for (int v = 0; v < 8; ++v) {
            const int j   = j0 + v + 8 * hi;
            const int idx = (i * L_DIM + j) * P_DIM + p0 + lane15;
            const float bv = __builtin_nontemporal_load(bias + idx);
            __builtin_nontemporal_store(acc[v] + bb + bv, out + idx);
        }
    }
}

// ---------------------------------------------------------------------------
extern "C" void kernel_launch(void* const* d_in, const int* in_sizes, int n_in,
                              void* d_out, int out_size, void* d_ws, size_t ws_size,
                              hipStream_t stream) {
    const float* seq  = (const float*)d_in[0];  // (1, 256, 256)
    const float* bias = (const float*)d_in[1];  // (1, 256, 256, 128)
    const float* W1   = (const float*)d_in[2];  // (32, 256)
    const float* b1   = (const float*)d_in[3];  // (32,)
    const float* W2   = (const float*)d_in[4];  // (128, 1024) == (4096, 32) K-major
    const float* b2   = (const float*)d_in[5];  // (128,)
    float* out = (float*)d_out;                 // (1, 256, 256, 128)

    float* X = (float*)d_ws;                    // 256*32   floats (32 KB)
    float* Y = X + L_DIM * M_DIM;               // 256*4096 floats (4 MB)

    opm_proj1<<<4,   256, 0, stream>>>(seq, W1, b1, X);
    opm_proj2<<<512, 256, 0, stream>>>(X, W2, Y);
    opm_outer<<<dim3(256, 2), 256, 0, stream>>>(X, Y, b2, bias, out);
}